// MultiLabelGHMCLossWithLogits_11123965296942
// MI455X (gfx1250) — compile-verified
//
#include <hip/hip_runtime.h>
#include <hip/hip_bf16.h>

#define GHM_BINS 30
#define COLS_PER_TILE 64
#define BLOCK_THREADS 256

typedef __attribute__((ext_vector_type(2))) float v2f;
typedef __attribute__((ext_vector_type(8))) float v8f;

// ---------------------------------------------------------------------------
// Wave-level sum via V_WMMA_F32_16X16X4_F32.
// A (16x4 f32): lane L, vgpr0 = s_L, vgpr1 = 0  ->  A[m,0]=s_m, A[m,2]=s_{m+16}
// B (4x16 f32): all ones (layout-independent).
// D[m,n] = s_m + s_{m+16}.  Lane L holds col N=L%16, rows M = r + 8*(L/16).
// sum of lane's 8 acc regs + same from lane L^16 == full wave sum.
// Requires EXEC all-ones: call with all 32 lanes active, no divergence.
// ---------------------------------------------------------------------------
__device__ __forceinline__ float wave_sum_wmma(float s) {
    v2f a; a.x = s;    a.y = 0.0f;
    v2f b; b.x = 1.0f; b.y = 1.0f;
    v8f c = {};
    c = __builtin_amdgcn_wmma_f32_16x16x4_f32(
            /*neg_a=*/false, a, /*neg_b=*/false, b,
            /*c_mod=*/(short)0, c, /*reuse_a=*/false, /*reuse_b=*/false);
    float h = c[0] + c[1] + c[2] + c[3] + c[4] + c[5] + c[6] + c[7];
    h += __shfl_xor(h, 16, 32);   // add the other half-row block
    return h;                     // every lane now holds the wave total
}

__device__ __forceinline__ int ghm_bin(float p, float t) {
    // g = |sigmoid(p) - t|, bin = clip(floor(g*BINS), 0, BINS-1)
    float sg = 1.0f / (1.0f + expf(-p));
    float g  = fabsf(sg - t);
    int b = (int)(g * (float)GHM_BINS);   // g >= 0, trunc == floor
    b = b < 0 ? 0 : b;
    b = b > (GHM_BINS - 1) ? (GHM_BINS - 1) : b;
    return b;
}

__device__ __forceinline__ float ghm_bce(float p, float t) {
    // logaddexp(0, p) - p*t  (stable form)
    return fmaxf(p, 0.0f) + log1pf(expf(-fabsf(p))) - p * t;
}

// ---------------------------------------------------------------------------
__global__ void ghm_zero_counts(unsigned int* __restrict__ counts, int n) {
    int i = blockIdx.x * blockDim.x + threadIdx.x;
    if (i < n) counts[i] = 0u;
}

// ---------------------------------------------------------------------------
// Histogram, vectorized: grid = (C/64, rowSplit), block = 256.
// Thread t: col-group = t%16 (4 consecutive cols, one float4/int4),
//           row offset = t/16, stepping 16 rows per iter -> GLOBAL_LOAD_B128.
// LDS hist [BINS][64], integer flush atomics (exact -> deterministic).
__global__ void ghm_hist_v4(const float* __restrict__ preds,
                            const int* __restrict__ targets,
                            unsigned int* __restrict__ counts,
                            int N, int C, int rowsPerBlock) {
    __shared__ unsigned int h[GHM_BINS * COLS_PER_TILE];
    for (int i = threadIdx.x; i < GHM_BINS * COLS_PER_TILE; i += blockDim.x) h[i] = 0u;
    __syncthreads();

    const int C4 = C >> 2;
    const int cg = threadIdx.x & 15;               // float4 column group in tile
    const int c4 = blockIdx.x * 16 + cg;           // global float4 column index
    const int clBase = cg << 2;                    // local column base (0..60)
    const int rowStart = blockIdx.y * rowsPerBlock;
    const int rowEnd   = min(rowStart + rowsPerBlock, N);
    const float4* __restrict__ p4 = (const float4*)preds;
    const int4*   __restrict__ t4 = (const int4*)targets;

    for (int r = rowStart + (threadIdx.x >> 4); r < rowEnd; r += 16) {
        size_t idx = (size_t)r * (size_t)C4 + (size_t)c4;
        float4 p = p4[idx];
        int4   t = t4[idx];
        atomicAdd(&h[ghm_bin(p.x, (float)t.x) * COLS_PER_TILE + clBase + 0], 1u);
        atomicAdd(&h[ghm_bin(p.y, (float)t.y) * COLS_PER_TILE + clBase + 1], 1u);
        atomicAdd(&h[ghm_bin(p.z, (float)t.z) * COLS_PER_TILE + clBase + 2], 1u);
        atomicAdd(&h[ghm_bin(p.w, (float)t.w) * COLS_PER_TILE + clBase + 3], 1u);
    }
    __syncthreads();

    for (int i = threadIdx.x; i < GHM_BINS * COLS_PER_TILE; i += blockDim.x) {
        unsigned int v = h[i];
        int b  = i >> 6;
        int cc = blockIdx.x * COLS_PER_TILE + (i & (COLS_PER_TILE - 1));
        if (v) atomicAdd(&counts[(size_t)b * C + cc], v);
    }
}

// Scalar fallback (C not divisible by 4).
__global__ void ghm_hist_s(const float* __restrict__ preds,
                           const int* __restrict__ targets,
                           unsigned int* __restrict__ counts,
                           int N, int C, int rowsPerBlock) {
    __shared__ unsigned int h[GHM_BINS * COLS_PER_TILE];
    for (int i = threadIdx.x; i < GHM_BINS * COLS_PER_TILE; i += blockDim.x) h[i] = 0u;
    __syncthreads();

    const int cl   = threadIdx.x & (COLS_PER_TILE - 1);
    const int col  = blockIdx.x * COLS_PER_TILE + cl;
    const bool act = (col < C);
    const int rowStart = blockIdx.y * rowsPerBlock;
    const int rowEnd   = min(rowStart + rowsPerBlock, N);

    for (int r = rowStart + (threadIdx.x >> 6); r < rowEnd; r += 4) {
        if (act) {
            size_t idx = (size_t)r * (size_t)C + (size_t)col;
            int b = ghm_bin(preds[idx], (float)targets[idx]);
            atomicAdd(&h[b * COLS_PER_TILE + cl], 1u);
        }
    }
    __syncthreads();

    for (int i = threadIdx.x; i < GHM_BINS * COLS_PER_TILE; i += blockDim.x) {
        unsigned int v = h[i];
        int b  = i >> 6;
        int cc = blockIdx.x * COLS_PER_TILE + (i & (COLS_PER_TILE - 1));
        if (v && cc < C) atomicAdd(&counts[(size_t)b * C + cc], v);
    }
}

// ---------------------------------------------------------------------------
// Per-column weight table: w[b,c] = (tot / (0.75*acc + 0.25*cnt)) / max(n_c,1)
__global__ void ghm_weights(const unsigned int* __restrict__ counts,
                            const float* __restrict__ acc_sum,
                            float* __restrict__ wtab,
                            int C, float tot) {
    int c = blockIdx.x * blockDim.x + threadIdx.x;
    if (c >= C) return;
    float accn[GHM_BINS];
    int n = 0;
    for (int b = 0; b < GHM_BINS; ++b) {
        float cnt = (float)counts[(size_t)b * C + c];
        accn[b] = 0.75f * acc_sum[(size_t)b * C + c] + 0.25f * cnt;
        n += (cnt >= 1.0f) ? 1 : 0;
    }
    float nn = fmaxf((float)n, 1.0f);
    for (int b = 0; b < GHM_BINS; ++b) {
        wtab[(size_t)b * C + c] = (tot / accn[b]) / nn;
    }
}

// ---------------------------------------------------------------------------
// Weighted-BCE partial sums, vectorized: grid = (C/64, rowSplit), block = 256.
__global__ void ghm_loss_v4(const float* __restrict__ preds,
                            const int* __restrict__ targets,
                            const float* __restrict__ wtab,
                            float* __restrict__ partials,
                            int N, int C, int rowsPerBlock) {
    const int C4 = C >> 2;
    const int cg = threadIdx.x & 15;
    const int c4 = blockIdx.x * 16 + cg;
    const int col = c4 << 2;                       // first of 4 global columns
    const int rowStart = blockIdx.y * rowsPerBlock;
    const int rowEnd   = min(rowStart + rowsPerBlock, N);
    const float4* __restrict__ p4 = (const float4*)preds;
    const int4*   __restrict__ t4 = (const int4*)targets;

    float s = 0.0f;
    for (int r = rowStart + (threadIdx.x >> 4); r < rowEnd; r += 16) {
        size_t idx = (size_t)r * (size_t)C4 + (size_t)c4;
        float4 p = p4[idx];
        int4   ti = t4[idx];
        float tx = (float)ti.x, ty = (float)ti.y, tz = (float)ti.z, tw = (float)ti.w;
        float wx = wtab[(size_t)ghm_bin(p.x, tx) * C + col + 0];
        float wy = wtab[(size_t)ghm_bin(p.y, ty) * C + col + 1];
        float wz = wtab[(size_t)ghm_bin(p.z, tz) * C + col + 2];
        float ww = wtab[(size_t)ghm_bin(p.w, tw) * C + col + 3];
        s = fmaf(wx, ghm_bce(p.x, tx), s);
        s = fmaf(wy, ghm_bce(p.y, ty), s);
        s = fmaf(wz, ghm_bce(p.z, tz), s);
        s = fmaf(ww, ghm_bce(p.w, tw), s);
    }

    // All 256 threads reach here together (EXEC all-ones for WMMA).
    float wsum = wave_sum_wmma(s);

    __shared__ float wpart[BLOCK_THREADS / 32];
    int wid = threadIdx.x >> 5;
    if ((threadIdx.x & 31) == 0) wpart[wid] = wsum;
    __syncthreads();
    if (threadIdx.x == 0) {
        float tot = 0.0f;
        for (int i = 0; i < BLOCK_THREADS / 32; ++i) tot += wpart[i];  // fixed order
        partials[(size_t)blockIdx.y * gridDim.x + blockIdx.x] = tot;
    }
}

// Scalar fallback (C not divisible by 4).
__global__ void ghm_loss_s(const float* __restrict__ preds,
                           const int* __restrict__ targets,
                           const float* __restrict__ wtab,
                           float* __restrict__ partials,
                           int N, int C, int rowsPerBlock) {
    const int cl   = threadIdx.x & (COLS_PER_TILE - 1);
    const int col  = blockIdx.x * COLS_PER_TILE + cl;
    const bool act = (col < C);
    const int rowStart = blockIdx.y * rowsPerBlock;
    const int rowEnd   = min(rowStart + rowsPerBlock, N);

    float s = 0.0f;
    for (int r = rowStart + (threadIdx.x >> 6); r < rowEnd; r += 4) {
        if (act) {
            size_t idx = (size_t)r * (size_t)C + (size_t)col;
            float p = preds[idx];
            float t = (float)targets[idx];
            float w = wtab[(size_t)ghm_bin(p, t) * C + col];
            s = fmaf(w, ghm_bce(p, t), s);
        }
    }

    float wsum = wave_sum_wmma(s);

    __shared__ float wpart[BLOCK_THREADS / 32];
    int wid = threadIdx.x >> 5;
    if ((threadIdx.x & 31) == 0) wpart[wid] = wsum;
    __syncthreads();
    if (threadIdx.x == 0) {
        float tot = 0.0f;
        for (int i = 0; i < BLOCK_THREADS / 32; ++i) tot += wpart[i];
        partials[(size_t)blockIdx.y * gridDim.x + blockIdx.x] = tot;
    }
}

// ---------------------------------------------------------------------------
// Deterministic single-block final reduction.
__global__ void ghm_finalize(const float* __restrict__ partials, int nPart,
                             float* __restrict__ out, float invNC) {
    __shared__ float sh[BLOCK_THREADS];
    float s = 0.0f;
    for (int i = threadIdx.x; i < nPart; i += BLOCK_THREADS) s += partials[i];
    sh[threadIdx.x] = s;
    __syncthreads();
    for (int k = BLOCK_THREADS / 2; k > 0; k >>= 1) {
        if (threadIdx.x < (unsigned)k) sh[threadIdx.x] += sh[threadIdx.x + k];
        __syncthreads();
    }
    if (threadIdx.x == 0) out[0] = sh[0] * invNC;   // LOSS_WEIGHT == 1.0
}

// ---------------------------------------------------------------------------
extern "C" void kernel_launch(void* const* d_in, const int* in_sizes, int n_in,
                              void* d_out, int out_size, void* d_ws, size_t ws_size,
                              hipStream_t stream) {
    const float* preds   = (const float*)d_in[0];
    const int*   targets = (const int*)d_in[1];
    const float* acc_sum = (const float*)d_in[2];

    const int BC = in_sizes[2];          // BINS * C
    const int C  = BC / GHM_BINS;        // 2048
    const int N  = in_sizes[0] / C;      // 8192

    // Workspace layout
    unsigned int* counts = (unsigned int*)d_ws;                          // BINS*C u32
    float* wtab     = (float*)((char*)d_ws + sizeof(unsigned int) * BC); // BINS*C f32
    float* partials = (float*)((char*)wtab + sizeof(float) * BC);

    const int colTiles = (C + COLS_PER_TILE - 1) / COLS_PER_TILE;  // 32
    const int rowSplit = 64;                                        // blocks per column tile
    const int rowsPerBlock = (N + rowSplit - 1) / rowSplit;         // 128
    const int nPart = colTiles * rowSplit;                          // 2048
    const bool vec = ((C & 3) == 0) && ((C % COLS_PER_TILE) == 0);

    ghm_zero_counts<<<(BC + BLOCK_THREADS - 1) / BLOCK_THREADS, BLOCK_THREADS, 0, stream>>>(
        counts, BC);

    if (vec) {
        ghm_hist_v4<<<dim3(colTiles, rowSplit), BLOCK_THREADS, 0, stream>>>(
            preds, targets, counts, N, C, rowsPerBlock);
    } else {
        ghm_hist_s<<<dim3(colTiles, rowSplit), BLOCK_THREADS, 0, stream>>>(
            preds, targets, counts, N, C, rowsPerBlock);
    }

    ghm_weights<<<(C + BLOCK_THREADS - 1) / BLOCK_THREADS, BLOCK_THREADS, 0, stream>>>(
        counts, acc_sum, wtab, C, (float)(N > 1 ? N : 1));

    if (vec) {
        ghm_loss_v4<<<dim3(colTiles, rowSplit), BLOCK_THREADS, 0, stream>>>(
            preds, targets, wtab, partials, N, C, rowsPerBlock);
    } else {
        ghm_loss_s<<<dim3(colTiles, rowSplit), BLOCK_THREADS, 0, stream>>>(
            preds, targets, wtab, partials, N, C, rowsPerBlock);
    }

    ghm_finalize<<<1, BLOCK_THREADS, 0, stream>>>(
        partials, nPart, (float*)d_out, 1.0f / ((float)N * (float)C));
}